// CompGCN_56298431316644
// MI455X (gfx1250) — compile-verified
//
#include <hip/hip_runtime.h>
#include <cstddef>

typedef __attribute__((ext_vector_type(2))) float v2f;
typedef __attribute__((ext_vector_type(8))) float v8f;

#define NUM_ENT 100000
#define NUM_REL 200
#define DD      200     // feature dim
#define DPAD    208     // padded column count / LDS stride
#define E_DIR   400000
#define BATCH   8192
#define BN_EPS  1e-5f
#define TILE_E  64      // edges (rows) per block tile
#define NT      13      // 13 * 16 = 208 column tiles
#define WP_SZ   (100 * 2 * DPAD)   // packed W: 100 fragment rows x 208 v2f

// ---------------------------------------------------------------------------
// Utility kernels
// ---------------------------------------------------------------------------
__global__ __launch_bounds__(256) void zero_kernel(float* __restrict__ p, size_t n) {
    size_t i = (size_t)blockIdx.x * blockDim.x + threadIdx.x;
    size_t stride = (size_t)gridDim.x * blockDim.x;
    for (; i < n; i += stride) p[i] = 0.0f;
}

__global__ __launch_bounds__(256) void deg_kernel(const int* __restrict__ src,
                                                  float* __restrict__ deg, int n) {
    int i = blockIdx.x * blockDim.x + threadIdx.x;
    if (i < n) atomicAdd(&deg[src[i]], 1.0f);
}

__global__ __launch_bounds__(256) void deginv_kernel(float* __restrict__ deg, int n) {
    int i = blockIdx.x * blockDim.x + threadIdx.x;
    if (i < n) {
        float d = deg[i];
        deg[i] = (d > 0.0f) ? rsqrtf(d) : 0.0f;
    }
}

// ---------------------------------------------------------------------------
// Pack a 200x200 weight into B-fragment layout, zero-padded to 208 columns.
// Fragment row f (0..99) holds pairs (w[2f][n], w[2f+1][n]) for n = 0..207,
// so a lane's per-K-step B fragment is ONE aligned v2f load:
//   b = ((const v2f*)wpack)[f * 208 + ncol]
// ---------------------------------------------------------------------------
__global__ __launch_bounds__(256) void pack_w_kernel(const float* __restrict__ w,
                                                     float* __restrict__ wpack) {
    int idx = blockIdx.x * blockDim.x + threadIdx.x;   // over 200*208 entries
    if (idx >= DD * DPAD) return;
    int k = idx / DPAD;
    int n = idx % DPAD;
    float v = (n < DD) ? w[(size_t)k * DD + n] : 0.0f;
    wpack[(size_t)(k >> 1) * (2 * DPAD) + 2 * n + (k & 1)] = v;
}

// ---------------------------------------------------------------------------
// Core 16x16 output-tile GEMM: acc = As[eL:eL+16][0:200] @ W  (K-loop of 4,
// V_WMMA_F32_16X16X4_F32). A from LDS (v2f), B from packed W (v2f, no guard).
// ---------------------------------------------------------------------------
__device__ __forceinline__ v8f tile_mm(const float (*As)[DPAD],
                                       const float* __restrict__ wpack,
                                       int eL, int ncol, int r16, int half) {
    v8f acc = {};
    const float* ap = &As[eL + r16][2 * half];
    const v2f*   bp = (const v2f*)wpack + (size_t)half * DPAD + ncol;
#pragma unroll 10
    for (int k0 = 0; k0 < DD; k0 += 4) {
        v2f a = *(const v2f*)ap;
        v2f b = *bp;
        acc = __builtin_amdgcn_wmma_f32_16x16x4_f32(
            false, a, false, b, (short)0, acc, false, false);
        ap += 4;            // next K-quad in this lane's A row
        bp += 2 * DPAD;     // advance two fragment rows (K += 4)
    }
    return acc;
}

// ---------------------------------------------------------------------------
// Fused edge-message kernel:
//   A[e][d] = x[src[e]][d] * rel[etype[e]][d]   (LDS stage)
//   msg = A @ W  via WMMA;  out[dst[e]][n] += msg * deg_inv[src]*deg_inv[dst]
// ---------------------------------------------------------------------------
__global__ __launch_bounds__(256) void edge_msg_kernel(
    const float* __restrict__ x,        // NUM_ENT x 200
    const float* __restrict__ rel,      // 401 x 200
    const float* __restrict__ wpack,    // packed 200x208 weight
    const int*   __restrict__ srcIdx,
    const int*   __restrict__ dstIdx,
    const int*   __restrict__ etype,
    const float* __restrict__ deg_inv,  // NUM_ENT
    float*       __restrict__ out,      // NUM_ENT x 200 accumulator
    int numEdges)
{
    __shared__ float As[TILE_E][DPAD];
    __shared__ float s_norm[TILE_E];
    __shared__ int   s_dst[TILE_E];

    const int tid  = threadIdx.x;
    const int base = blockIdx.x * TILE_E;

    // ---- stage A tile into LDS (4 threads per edge row) ----
    {
        int e   = tid >> 2;        // 0..63
        int sub = tid & 3;
        int eg  = base + e;
        bool valid = (eg < numEdges);
        int src = 0, et = 0;
        if (valid) { src = srcIdx[eg]; et = etype[eg]; }
        const float* xrow = x   + (size_t)src * DD;
        const float* rrow = rel + (size_t)et  * DD;
        for (int d = sub; d < DD; d += 4)
            As[e][d] = valid ? (xrow[d] * rrow[d]) : 0.0f;
        if (sub == 0) {
            int dst = valid ? dstIdx[eg] : 0;
            s_dst[e]  = dst;
            s_norm[e] = valid ? (deg_inv[src] * deg_inv[dst]) : 0.0f;
        }
    }
    __syncthreads();

    const int wave = tid >> 5;
    const int lane = tid & 31;
    const int r16  = lane & 15;
    const int half = lane >> 4;

    for (int t = wave; t < 4 * NT; t += 8) {
        int esub = t / NT;
        int nt   = t % NT;
        int ncol = nt * 16 + r16;
        int eL   = esub * 16;

        v8f acc = tile_mm(As, wpack, eL, ncol, r16, half);

        if (ncol < DD) {
#pragma unroll
            for (int i = 0; i < 8; ++i) {
                int eIdx = eL + i + 8 * half;
                float v  = acc[i] * s_norm[eIdx];
                atomicAdd(&out[(size_t)s_dst[eIdx] * DD + ncol], v);
            }
        }
    }
}

// ---------------------------------------------------------------------------
// Self-loop kernel: out[r][:] += (x[r][:] * rel[400][:]) @ W
// ---------------------------------------------------------------------------
__global__ __launch_bounds__(256) void loop_msg_kernel(
    const float* __restrict__ x,
    const float* __restrict__ rel,      // row 400 = loop relation
    const float* __restrict__ wpack,
    float*       __restrict__ out,
    int numRows)
{
    __shared__ float As[TILE_E][DPAD];

    const int tid  = threadIdx.x;
    const int base = blockIdx.x * TILE_E;

    {
        int e   = tid >> 2;
        int sub = tid & 3;
        int rg  = base + e;
        bool valid = (rg < numRows);
        const float* xrow = x + (size_t)(valid ? rg : 0) * DD;
        const float* rrow = rel + (size_t)(2 * NUM_REL) * DD;
        for (int d = sub; d < DD; d += 4)
            As[e][d] = valid ? (xrow[d] * rrow[d]) : 0.0f;
    }
    __syncthreads();

    const int wave = tid >> 5;
    const int lane = tid & 31;
    const int r16  = lane & 15;
    const int half = lane >> 4;

    for (int t = wave; t < 4 * NT; t += 8) {
        int esub = t / NT;
        int nt   = t % NT;
        int ncol = nt * 16 + r16;
        int eL   = esub * 16;

        v8f acc = tile_mm(As, wpack, eL, ncol, r16, half);

        if (ncol < DD) {
#pragma unroll
            for (int i = 0; i < 8; ++i) {
                int rg = base + eL + i + 8 * half;
                if (rg < numRows)
                    atomicAdd(&out[(size_t)rg * DD + ncol], acc[i]);
            }
        }
    }
}

// ---------------------------------------------------------------------------
// Rel GEMM: relOut[0:400][:] = (relIn @ W)[0:400][:]   (401 input rows)
// ---------------------------------------------------------------------------
__global__ __launch_bounds__(256) void relgemm_kernel(
    const float* __restrict__ relIn,   // 401 x 200
    const float* __restrict__ wpack,
    float*       __restrict__ relOut,  // 400 x 200
    int numRowsIn, int numRowsOut)
{
    __shared__ float As[TILE_E][DPAD];

    const int tid  = threadIdx.x;
    const int base = blockIdx.x * TILE_E;

    {
        int e   = tid >> 2;
        int sub = tid & 3;
        int rg  = base + e;
        bool valid = (rg < numRowsIn);
        const float* arow = relIn + (size_t)(valid ? rg : 0) * DD;
        for (int d = sub; d < DD; d += 4)
            As[e][d] = valid ? arow[d] : 0.0f;
    }
    __syncthreads();

    const int wave = tid >> 5;
    const int lane = tid & 31;
    const int r16  = lane & 15;
    const int half = lane >> 4;

    for (int t = wave; t < 4 * NT; t += 8) {
        int esub = t / NT;
        int nt   = t % NT;
        int ncol = nt * 16 + r16;
        int eL   = esub * 16;

        v8f acc = tile_mm(As, wpack, eL, ncol, r16, half);

        if (ncol < DD) {
#pragma unroll
            for (int i = 0; i < 8; ++i) {
                int rg = base + eL + i + 8 * half;
                if (rg < numRowsOut)
                    relOut[(size_t)rg * DD + ncol] = acc[i];
            }
        }
    }
}

// ---------------------------------------------------------------------------
// BatchNorm stats / apply / gather
// ---------------------------------------------------------------------------
__global__ __launch_bounds__(256) void bn_stats_kernel(
    const float* __restrict__ acc, float* __restrict__ mu,
    float* __restrict__ var, int nrows)
{
    __shared__ float ssum[256];
    __shared__ float ssq[256];
    const int col = blockIdx.x;
    float s = 0.0f, q = 0.0f;
    for (int r = threadIdx.x; r < nrows; r += 256) {
        float y = acc[(size_t)r * DD + col] * (1.0f / 3.0f);
        s += y;
        q += y * y;
    }
    ssum[threadIdx.x] = s;
    ssq[threadIdx.x]  = q;
    __syncthreads();
    for (int off = 128; off > 0; off >>= 1) {
        if (threadIdx.x < off) {
            ssum[threadIdx.x] += ssum[threadIdx.x + off];
            ssq[threadIdx.x]  += ssq[threadIdx.x + off];
        }
        __syncthreads();
    }
    if (threadIdx.x == 0) {
        float m = ssum[0] / (float)nrows;
        mu[col]  = m;
        var[col] = ssq[0] / (float)nrows - m * m;
    }
}

__global__ __launch_bounds__(256) void bn_apply_kernel(
    float* __restrict__ acc, const float* __restrict__ mu,
    const float* __restrict__ var, size_t n)
{
    size_t i = (size_t)blockIdx.x * blockDim.x + threadIdx.x;
    size_t stride = (size_t)gridDim.x * blockDim.x;
    for (; i < n; i += stride) {
        int col = (int)(i % DD);
        float y = acc[i] * (1.0f / 3.0f);
        acc[i] = tanhf((y - mu[col]) * rsqrtf(var[col] + BN_EPS));
    }
}

__global__ __launch_bounds__(256) void gather_kernel(
    const float* __restrict__ x, const int* __restrict__ org,
    float* __restrict__ out, int nrows)
{
    size_t i = (size_t)blockIdx.x * blockDim.x + threadIdx.x;
    size_t n = (size_t)nrows * DD;
    size_t stride = (size_t)gridDim.x * blockDim.x;
    for (; i < n; i += stride) {
        int r = (int)(i / DD);
        int c = (int)(i % DD);
        out[i] = x[(size_t)org[r] * DD + c];
    }
}

// ---------------------------------------------------------------------------
// Host-side launch
// ---------------------------------------------------------------------------
extern "C" void kernel_launch(void* const* d_in, const int* in_sizes, int n_in,
                              void* d_out, int out_size, void* d_ws, size_t ws_size,
                              hipStream_t stream) {
    const float* x0        = (const float*)d_in[0];
    const float* init_rel  = (const float*)d_in[1];
    const float* w_in1     = (const float*)d_in[2];
    const float* w_out1    = (const float*)d_in[3];
    const float* w_loop1   = (const float*)d_in[4];
    const float* w_rel1    = (const float*)d_in[5];
    const float* loop_rel1 = (const float*)d_in[6];
    const float* w_in2     = (const float*)d_in[7];
    const float* w_out2    = (const float*)d_in[8];
    const float* w_loop2   = (const float*)d_in[9];
    // d_in[10] = w_rel2 (unused: layer-2 rel output is discarded)
    const float* loop_rel2 = (const float*)d_in[11];
    const int*   ei        = (const int*)d_in[12];
    const int*   et        = (const int*)d_in[13];
    const int*   org       = (const int*)d_in[14];

    const int E = E_DIR;
    const int* src_in  = ei;
    const int* dst_in  = ei + 2 * E_DIR;
    const int* src_out = ei + E_DIR;
    const int* dst_out = ei + 3 * E_DIR;
    const int* et_in   = et;
    const int* et_out  = et + E_DIR;

    // workspace layout (floats)
    float* wsf      = (float*)d_ws;
    float* deg_in   = wsf;                                // NUM_ENT
    float* deg_out  = deg_in  + NUM_ENT;                  // NUM_ENT
    float* rel_a    = deg_out + NUM_ENT;                  // 401*200
    float* rel_b    = rel_a   + (2 * NUM_REL + 1) * DD;   // 401*200
    float* mu       = rel_b   + (2 * NUM_REL + 1) * DD;   // 256
    float* var      = mu + 256;                           // 256
    float* wp_in1   = var + 256;                          // 7 packed weights
    float* wp_out1  = wp_in1  + WP_SZ;
    float* wp_loop1 = wp_out1 + WP_SZ;
    float* wp_rel1  = wp_loop1 + WP_SZ;
    float* wp_in2   = wp_rel1 + WP_SZ;
    float* wp_out2  = wp_in2  + WP_SZ;
    float* wp_loop2 = wp_out2 + WP_SZ;
    float* X1       = wp_loop2 + WP_SZ;                   // NUM_ENT*200
    float* X2       = (float*)d_out + (size_t)BATCH * DD; // layer-2 out

    const size_t NXD = (size_t)NUM_ENT * DD;
    const int edgeBlocks = (E + TILE_E - 1) / TILE_E;
    const int entBlocks  = (NUM_ENT + TILE_E - 1) / TILE_E;
    const int relBlocks  = (2 * NUM_REL + 1 + TILE_E - 1) / TILE_E;
    const int packBlocks = (DD * DPAD + 255) / 256;

    // ---- pack weights into fragment-ready zero-padded layout ----
    pack_w_kernel<<<packBlocks, 256, 0, stream>>>(w_in1,   wp_in1);
    pack_w_kernel<<<packBlocks, 256, 0, stream>>>(w_out1,  wp_out1);
    pack_w_kernel<<<packBlocks, 256, 0, stream>>>(w_loop1, wp_loop1);
    pack_w_kernel<<<packBlocks, 256, 0, stream>>>(w_rel1,  wp_rel1);
    pack_w_kernel<<<packBlocks, 256, 0, stream>>>(w_in2,   wp_in2);
    pack_w_kernel<<<packBlocks, 256, 0, stream>>>(w_out2,  wp_out2);
    pack_w_kernel<<<packBlocks, 256, 0, stream>>>(w_loop2, wp_loop2);

    // ---- degrees (same for both layers) ----
    zero_kernel<<<512, 256, 0, stream>>>(deg_in, (size_t)2 * NUM_ENT);
    deg_kernel<<<(E + 255) / 256, 256, 0, stream>>>(src_in,  deg_in,  E);
    deg_kernel<<<(E + 255) / 256, 256, 0, stream>>>(src_out, deg_out, E);
    deginv_kernel<<<(2 * NUM_ENT + 255) / 256, 256, 0, stream>>>(deg_in, 2 * NUM_ENT);

    // ---- layer 1 rel table: concat(init_rel, loop_rel1) ----
    hipMemcpyAsync(rel_a, init_rel, (size_t)2 * NUM_REL * DD * sizeof(float),
                   hipMemcpyDeviceToDevice, stream);
    hipMemcpyAsync(rel_a + (size_t)2 * NUM_REL * DD, loop_rel1, DD * sizeof(float),
                   hipMemcpyDeviceToDevice, stream);

    // ---- layer 1 ----
    zero_kernel<<<4096, 256, 0, stream>>>(X1, NXD);
    edge_msg_kernel<<<edgeBlocks, 256, 0, stream>>>(x0, rel_a, wp_in1,  src_in,  dst_in,  et_in,  deg_in,  X1, E);
    edge_msg_kernel<<<edgeBlocks, 256, 0, stream>>>(x0, rel_a, wp_out1, src_out, dst_out, et_out, deg_out, X1, E);
    loop_msg_kernel<<<entBlocks, 256, 0, stream>>>(x0, rel_a, wp_loop1, X1, NUM_ENT);
    bn_stats_kernel<<<DD, 256, 0, stream>>>(X1, mu, var, NUM_ENT);
    bn_apply_kernel<<<4096, 256, 0, stream>>>(X1, mu, var, NXD);

    // ---- layer 2 rel table: concat((rel_a @ w_rel1)[:400], loop_rel2) ----
    relgemm_kernel<<<relBlocks, 256, 0, stream>>>(rel_a, wp_rel1, rel_b,
                                                  2 * NUM_REL + 1, 2 * NUM_REL);
    hipMemcpyAsync(rel_b + (size_t)2 * NUM_REL * DD, loop_rel2, DD * sizeof(float),
                   hipMemcpyDeviceToDevice, stream);

    // ---- layer 2 (accumulate directly into d_out's x region) ----
    zero_kernel<<<4096, 256, 0, stream>>>(X2, NXD);
    edge_msg_kernel<<<edgeBlocks, 256, 0, stream>>>(X1, rel_b, wp_in2,  src_in,  dst_in,  et_in,  deg_in,  X2, E);
    edge_msg_kernel<<<edgeBlocks, 256, 0, stream>>>(X1, rel_b, wp_out2, src_out, dst_out, et_out, deg_out, X2, E);
    loop_msg_kernel<<<entBlocks, 256, 0, stream>>>(X1, rel_b, wp_loop2, X2, NUM_ENT);
    bn_stats_kernel<<<DD, 256, 0, stream>>>(X2, mu, var, NUM_ENT);
    bn_apply_kernel<<<4096, 256, 0, stream>>>(X2, mu, var, NXD);

    // ---- org gather ----
    gather_kernel<<<1600, 256, 0, stream>>>(X2, org, (float*)d_out, BATCH);
}